// SCF_4269197492539
// MI455X (gfx1250) — compile-verified
//
#include <hip/hip_runtime.h>
#include <hip/hip_bf16.h>

#define N_USERS  200000
#define N_ITEMS  100000
#define DIM      64
#define N_EDGES  1200000
#define N_TOTAL  (N_USERS + N_ITEMS)
#define NTILES   (N_TOTAL / 16)          // 18750 exactly

typedef float v2f __attribute__((ext_vector_type(2)));
typedef float v8f __attribute__((ext_vector_type(8)));

// ---------------- Pass 1: zero the aggregation scratch (N x 64 f32) -------
__global__ __launch_bounds__(256) void SCF_zero_ws(float* __restrict__ ws) {
    size_t i = (size_t)blockIdx.x * 256u + threadIdx.x;   // grid sized exactly
    float4 z = {0.f, 0.f, 0.f, 0.f};
    ((float4*)ws)[i] = z;
}

// ---------------- Pass 2: edge gather-scale + atomic scatter-add ----------
// One wave per edge: lane p handles columns {2p, 2p+1}. emb-row reads and
// atomic bursts are 256B contiguous per wave.
__global__ __launch_bounds__(256) void SCF_scatter_edges(
        const int*   __restrict__ rows,
        const int*   __restrict__ cols,
        const float* __restrict__ vals,
        const float* __restrict__ uemb,
        const float* __restrict__ iemb,
        float*       __restrict__ agg) {
    unsigned tid = blockIdx.x * 256u + threadIdx.x;
    unsigned e   = tid >> 5;
    unsigned p   = (tid & 31u) * 2u;
    if (e >= N_EDGES) return;
    int   r = rows[e];
    int   c = cols[e];
    float v = vals[e];
    const float* src = (c < N_USERS) ? (uemb + (size_t)c * DIM)
                                     : (iemb + (size_t)(c - N_USERS) * DIM);
    v2f x = *(const v2f*)(src + p);
    float* dst = agg + (size_t)r * DIM + p;
    unsafeAtomicAdd(dst + 0, v * x.x);   // global_atomic_add_f32 (no CAS loop)
    unsafeAtomicAdd(dst + 1, v * x.y);
}

// ---------------- Pass 3: fused X = 2*emb - agg, H = sigmoid(X @ filt) ----
// One wave per 16-row tile. V_WMMA_F32_16X16X4_F32, K=64 => 16 k-steps per
// 16x16 output tile, 4 output tiles (N=64). f32 end-to-end (exact).
__global__ __launch_bounds__(256) void SCF_gcn_wmma(
        const float* __restrict__ uemb,
        const float* __restrict__ iemb,
        const float* __restrict__ filt,
        const float* __restrict__ agg,
        float*       __restrict__ out) {
    const unsigned lane = threadIdx.x & 31u;
    const unsigned wave = threadIdx.x >> 5;
    const unsigned tile = blockIdx.x * 8u + wave;
    if (tile >= NTILES) return;                  // wave-uniform: EXEC all-1s below

    const unsigned m    = lane & 15u;            // row within tile (A) / col (B,D)
    const unsigned koff = (lane >> 4) * 2u;      // 0 for lanes 0-15, 2 for 16-31
    const unsigned row  = tile * 16u + m;
    const float* embRow = (row < N_USERS) ? (uemb + (size_t)row * DIM)
                                          : (iemb + (size_t)(row - N_USERS) * DIM);
    const float* aggRow = agg + (size_t)row * DIM;
    float*       outRow = out + (size_t)row * 128u;

    // Build A fragments (ISA 16x4 f32 layout) and emit left half of output.
    v2f a[16];
#pragma unroll
    for (int k = 0; k < 16; ++k) {
        unsigned c = 4u * k + koff;
        v2f ev = *(const v2f*)(embRow + c);
        v2f gv = *(const v2f*)(aggRow + c);
        v2f xv;
        xv.x = 2.0f * ev.x - gv.x;
        xv.y = 2.0f * ev.y - gv.y;
        a[k] = xv;
        *(v2f*)(outRow + c) = ev;                // out[:, 0:64] = emb
    }

    const unsigned n     = m;                    // B/D column index
    const unsigned mbase = (lane >> 4) * 8u;     // D: M = mbase + vgpr
#pragma unroll
    for (int j = 0; j < 4; ++j) {                // 16-col output tiles
        v8f acc = {0.f,0.f,0.f,0.f,0.f,0.f,0.f,0.f};
#pragma unroll
        for (int k = 0; k < 16; ++k) {
            unsigned krow = 4u * k + koff;       // B 4x16 layout mirrors A
            v2f b;
            b.x = filt[(size_t)(krow    ) * DIM + j * 16 + n];
            b.y = filt[(size_t)(krow + 1) * DIM + j * 16 + n];
            acc = __builtin_amdgcn_wmma_f32_16x16x4_f32(
                    false, a[k], false, b, (short)0, acc, false, false);
        }
#pragma unroll
        for (int r = 0; r < 8; ++r) {
            float h = acc[r];
            h = 1.0f / (1.0f + __expf(-h));      // sigmoid
            unsigned orow = tile * 16u + mbase + r;
            out[(size_t)orow * 128u + 64u + j * 16u + n] = h;
        }
    }
}

extern "C" void kernel_launch(void* const* d_in, const int* in_sizes, int n_in,
                              void* d_out, int out_size, void* d_ws, size_t ws_size,
                              hipStream_t stream) {
    const int*   adj_rows = (const int*)d_in[0];
    const int*   adj_cols = (const int*)d_in[1];
    const float* adj_vals = (const float*)d_in[2];
    const float* user_emb = (const float*)d_in[3];
    const float* item_emb = (const float*)d_in[4];
    const float* filt     = (const float*)d_in[5];
    float* out = (float*)d_out;
    float* agg = (float*)d_ws;                   // N_TOTAL * 64 f32 = 76.8 MB

    // Pass 1: zero agg (exactly 4.8M float4 -> 18750 blocks)
    SCF_zero_ws<<<(N_TOTAL * DIM / 4 + 255) / 256, 256, 0, stream>>>(agg);

    // Pass 2: scatter (1.2M edges * 32 lanes = 150000 blocks)
    SCF_scatter_edges<<<(N_EDGES * 32 + 255) / 256, 256, 0, stream>>>(
        adj_rows, adj_cols, adj_vals, user_emb, item_emb, agg);

    // Pass 3: fused WMMA + sigmoid + pack (18750 tiles, 8 waves/block)
    SCF_gcn_wmma<<<(NTILES + 7) / 8, 256, 0, stream>>>(
        user_emb, item_emb, filt, agg, out);
}